// OnlineDictionaryLearning_56573309224025
// MI455X (gfx1250) — compile-verified
//
#include <hip/hip_runtime.h>
#include <math.h>

typedef __attribute__((ext_vector_type(2))) float v2f;
typedef __attribute__((ext_vector_type(8))) float v8f;

#define BATCH     16384
#define DIM       512
#define NATOMS    1024
#define SPARSITY  5
#define XS_PITCH  516   // 516 % 64 == 4  -> conflict-free b64 fragment reads

// ---------------------------------------------------------------------------
// Row-gather A fragment for V_WMMA_F32_16X16X4_F32 from a row-major [M x K]
// matrix (used only in the tiny Gram kernel): lanes 0-15 hold K={k,k+1},
// lanes 16-31 hold K={k+2,k+3}, for row (lane & 15).
// ---------------------------------------------------------------------------
__device__ __forceinline__ v2f afrag_rows(const float* __restrict__ base, int ld,
                                          int row0, int lane, int k) {
    int r  = lane & 15;
    int kk = k + ((lane >> 4) << 1);
    const float* p = base + (size_t)(row0 + r) * ld + kk;
    v2f v; v.x = p[0]; v.y = p[1];
    return v;
}

// ---------------------------------------------------------------------------
// Coalesced B fragment from the TRANSPOSED dictionary DnT [K=512 x N=1024]:
// lane group reads one row of DnT at consecutive columns -> coalesced b32s.
// ---------------------------------------------------------------------------
__device__ __forceinline__ v2f bfrag_T(const float* __restrict__ DnT,
                                       int n, int lane, int k) {
    int koff = (lane >> 4) << 1;
    v2f b;
    b.x = DnT[(size_t)(k + koff)     * NATOMS + n];
    b.y = DnT[(size_t)(k + koff + 1) * NATOMS + n];
    return b;
}

// ---------------------------------------------------------------------------
// Kernel 1: row-normalize the dictionary; emit Dn (row-major) and DnT.
// ---------------------------------------------------------------------------
__global__ __launch_bounds__(256) void normalize_rows(const float* __restrict__ dict,
                                                      float* __restrict__ Dn,
                                                      float* __restrict__ DnT) {
    int lane = threadIdx.x & 31;
    int wave = threadIdx.x >> 5;
    int row  = blockIdx.x * 8 + wave;
    if (row >= NATOMS) return;
    const float* src = dict + (size_t)row * DIM;
    float ss = 0.0f;
#pragma unroll
    for (int i = 0; i < DIM / 32; ++i) {
        float v = src[i * 32 + lane];
        ss += v * v;
    }
#pragma unroll
    for (int off = 16; off > 0; off >>= 1)
        ss += __shfl_xor(ss, off, 32);
    float inv = 1.0f / sqrtf(ss);
    float* dst = Dn + (size_t)row * DIM;
#pragma unroll
    for (int i = 0; i < DIM / 32; ++i) {
        int   c = i * 32 + lane;
        float v = src[c] * inv;
        dst[c] = v;
        DnT[(size_t)c * NATOMS + row] = v;   // one-time 2 MB transpose
    }
}

// ---------------------------------------------------------------------------
// Kernel 2: Gram = Dn @ Dn^T (1024x1024, K=512). One 16x16 tile per wave.
// ---------------------------------------------------------------------------
__global__ __launch_bounds__(256) void gram_wmma(const float* __restrict__ Dn,
                                                 const float* __restrict__ DnT,
                                                 float* __restrict__ Gram) {
    int lane = threadIdx.x & 31;
    int wave = threadIdx.x >> 5;
    int tile = blockIdx.x * 8 + wave;       // 0 .. 4095
    int tm   = (tile >> 6) << 4;
    int tn   = (tile & 63) << 4;

    int n = tn + (lane & 15);
    v8f acc = {};
    for (int k = 0; k < DIM; k += 4) {
        v2f a = afrag_rows(Dn, DIM, tm, lane, k);
        v2f b = bfrag_T(DnT, n, lane, k);
        acc = __builtin_amdgcn_wmma_f32_16x16x4_f32(false, a, false, b,
                                                    (short)0, acc, false, false);
    }
    int row0 = tm + ((lane >> 4) << 3);
    int col  = tn + (lane & 15);
#pragma unroll
    for (int r = 0; r < 8; ++r)
        Gram[(size_t)(row0 + r) * NATOMS + col] = acc[r];
}

// ---------------------------------------------------------------------------
// Kernel 3 (fused): per 16-sample block
//   (a) stage x tile in LDS (coalesced), c0 = x_tile @ Dn^T via WMMA -> LDS
//   (b) batch-OMP per sample using Gram (L2 resident)
//   (c) scatter codes + write reconstruction
// ---------------------------------------------------------------------------
__global__ __launch_bounds__(256) void omp_fused(const float* __restrict__ x,
                                                 const float* __restrict__ Dn,
                                                 const float* __restrict__ DnT,
                                                 const float* __restrict__ Gram,
                                                 float* __restrict__ codes,
                                                 float* __restrict__ recon) {
    __shared__ float xs[16 * XS_PITCH];     // 33 KB, padded against bank conflicts
    __shared__ float c0s[16][NATOMS];       // 64 KB

    int lane = threadIdx.x & 31;
    int wave = threadIdx.x >> 5;
    int s0   = blockIdx.x * 16;

    // -------- stage x tile (coalesced) --------------------------------------
    for (int idx = threadIdx.x; idx < 16 * DIM; idx += 256) {
        int r = idx >> 9;
        int c = idx & (DIM - 1);
        xs[r * XS_PITCH + c] = x[(size_t)(s0 + r) * DIM + c];
    }
    __syncthreads();

    // -------- (a) GEMM: wave w -> 128 columns [w*128, w*128+128) ------------
    int n0   = wave * 128;
    int r    = lane & 15;
    int koff = (lane >> 4) << 1;
    v8f acc[8] = {};
    for (int k = 0; k < DIM; k += 4) {
        v2f a = *(const v2f*)&xs[r * XS_PITCH + k + koff];   // ds_load_b64, conflict-free
#pragma unroll
        for (int c = 0; c < 8; ++c) {
            v2f b = bfrag_T(DnT, n0 + c * 16 + (lane & 15), lane, k);
            acc[c] = __builtin_amdgcn_wmma_f32_16x16x4_f32(false, a, false, b,
                                                           (short)0, acc[c],
                                                           false, false);
        }
    }
    int rbase = (lane >> 4) << 3;
#pragma unroll
    for (int c = 0; c < 8; ++c)
#pragma unroll
        for (int rr = 0; rr < 8; ++rr)
            c0s[rbase + rr][n0 + c * 16 + (lane & 15)] = acc[c][rr];

    __syncthreads();

    // -------- (b) batch-OMP: wave w handles samples 2w, 2w+1 ---------------
    for (int ss = 0; ss < 2; ++ss) {
        int s = wave * 2 + ss;

        float c0loc[NATOMS / 32];
#pragma unroll
        for (int i = 0; i < NATOMS / 32; ++i)
            c0loc[i] = c0s[s][i * 32 + lane];

        int   sel[SPARSITY];
        float cf[SPARSITY];

        for (int t = 0; t < SPARSITY; ++t) {
            float best = -1.0f;
            int   bidx = NATOMS;
            for (int i = 0; i < NATOMS / 32; ++i) {
                int   k    = i * 32 + lane;
                float corr = c0loc[i];
                for (int j = 0; j < t; ++j)
                    corr -= cf[j] * Gram[(size_t)sel[j] * NATOMS + k];
                float av = fabsf(corr);
                if (av > best || (av == best && k < bidx)) { best = av; bidx = k; }
            }
#pragma unroll
            for (int off = 16; off > 0; off >>= 1) {
                float ob = __shfl_xor(best, off, 32);
                int   oi = __shfl_xor(bidx, off, 32);
                if (ob > best || (ob == best && oi < bidx)) { best = ob; bidx = oi; }
            }
            sel[t] = bidx;

            // tiny (t+1)x(t+1) normal-equations solve, redundant on all lanes
            float G[SPARSITY][SPARSITY];
            float rhs[SPARSITY];
            for (int i = 0; i <= t; ++i) {
                rhs[i] = c0s[s][sel[i]];
                for (int j = 0; j <= t; ++j)
                    G[i][j] = Gram[(size_t)sel[i] * NATOMS + sel[j]];
            }
            for (int p = 0; p <= t; ++p) {               // Gauss-Jordan
                float inv = 1.0f / G[p][p];
                for (int j = p; j <= t; ++j) G[p][j] *= inv;
                rhs[p] *= inv;
                for (int i2 = 0; i2 <= t; ++i2) {
                    if (i2 == p) continue;
                    float f = G[i2][p];
                    for (int j = p; j <= t; ++j) G[i2][j] -= f * G[p][j];
                    rhs[i2] -= f * rhs[p];
                }
            }
            for (int i = 0; i <= t; ++i) cf[i] = rhs[i];
        }

        // -------- (c) outputs (all coalesced) -------------------------------
        size_t crow = (size_t)(s0 + s) * NATOMS;
#pragma unroll
        for (int i = 0; i < NATOMS / 32; ++i)
            codes[crow + i * 32 + lane] = 0.0f;
        if (lane == 0) {
            for (int j = 0; j < SPARSITY; ++j)   // sequential: last write wins
                codes[crow + sel[j]] = cf[j];
        }

        size_t rrow = (size_t)(s0 + s) * DIM;
#pragma unroll
        for (int i = 0; i < DIM / 32; ++i) {
            int   col = i * 32 + lane;
            float v   = 0.0f;
            for (int j = 0; j < SPARSITY; ++j)
                v += cf[j] * Dn[(size_t)sel[j] * DIM + col];
            recon[rrow + col] = v;
        }
    }
}

// ---------------------------------------------------------------------------
extern "C" void kernel_launch(void* const* d_in, const int* in_sizes, int n_in,
                              void* d_out, int out_size, void* d_ws, size_t ws_size,
                              hipStream_t stream) {
    (void)in_sizes; (void)n_in; (void)out_size; (void)ws_size;

    const float* x    = (const float*)d_in[0];      // (16384, 512)
    const float* dict = (const float*)d_in[1];      // (1024, 512)

    float* out   = (float*)d_out;
    float* codes = out;                              // (16384, 1024)
    float* recon = out + (size_t)BATCH * NATOMS;     // (16384, 512)

    float* Dn   = (float*)d_ws;                      // 1024*512   = 2 MB
    float* DnT  = Dn  + (size_t)NATOMS * DIM;        // 512*1024   = 2 MB
    float* Gram = DnT + (size_t)DIM * NATOMS;        // 1024*1024  = 4 MB

    normalize_rows<<<NATOMS / 8, 256, 0, stream>>>(dict, Dn, DnT);
    gram_wmma<<<(64 * 64) / 8, 256, 0, stream>>>(Dn, DnT, Gram);
    omp_fused<<<BATCH / 16, 256, 0, stream>>>(x, Dn, DnT, Gram, codes, recon);
}